// RandomAmplitudeModulationAugment_28870770163738
// MI455X (gfx1250) — compile-verified
//
#include <hip/hip_runtime.h>
#include <cmath>

#define T_LEN    160000
#define N_CH     256
#define K_CHUNKS 256
#define L_CHUNK  625      // K_CHUNKS * L_CHUNK == T_LEN
#define TILE_T   1024     // mod samples staged in LDS per block
#define ROWS     8        // channels processed per block

typedef float v4f __attribute__((ext_vector_type(4)));
typedef unsigned int v4u __attribute__((ext_vector_type(4)));
typedef int v4i __attribute__((ext_vector_type(4)));
typedef int v8i __attribute__((ext_vector_type(8)));

// ---------------------------------------------------------------------------
// out[c,t] = x[c,t] * mod[t].  Streaming, HBM-bound (~14 us floor at 23.3TB/s).
// The mod tile is staged into LDS by the Tensor Data Mover (wave 0 issues a
// 1-D descriptor, waits TENSORcnt, then the block barriers).  x/out use
// non-temporal hints so the one-touch 163MB stream doesn't evict the 640KB
// broadcast mod vector from the 192MB L2.
// ---------------------------------------------------------------------------
__global__ void modmul_kernel(const float* __restrict__ x,
                              const float* __restrict__ mod,
                              float* __restrict__ out)
{
    __shared__ __align__(16) float smod[TILE_T];
    const int tstart = blockIdx.x * TILE_T;

#if defined(__HIP_DEVICE_COMPILE__) && __has_builtin(__builtin_amdgcn_tensor_load_to_lds)
    if (threadIdx.x < 32) {  // wave 0 issues the TDM transfer
        unsigned long long ga = (unsigned long long)(mod + tstart);
        unsigned int galo = (unsigned int)__builtin_amdgcn_readfirstlane((int)(ga & 0xFFFFFFFFull));
        unsigned int gahi = (unsigned int)__builtin_amdgcn_readfirstlane((int)(ga >> 32));
        unsigned int dim0 = (unsigned int)__builtin_amdgcn_readfirstlane(T_LEN - tstart);
        // LDS byte offset of smod: low 32 bits of the generic address
        unsigned int ldsoff = (unsigned int)__builtin_amdgcn_readfirstlane(
            (int)(unsigned int)(unsigned long long)(uintptr_t)&smod[0]);

        // D# group 0: count=1 | lds_addr | global_addr | type=2 ("image")
        v4u g0;
        g0.x = 1u;                                   // count=1, user mode
        g0.y = ldsoff;                               // lds_addr (bytes)
        g0.z = galo;                                 // global_addr[31:0]
        g0.w = (gahi & 0x01FFFFFFu) | (2u << 30);    // global_addr[56:32] | type=2

        // D# group 1: data_size=4B, tensor_dim0 = remaining elems (OOB -> 0),
        // tensor_dim1=1, tile = TILE_T x 1, stride0 = TILE_T.
        v8i g1;
        g1[0] = (int)(2u << 16);                         // data_size=2 (4B); wg_mask=0
        g1[1] = (int)((dim0 & 0xFFFFu) << 16);           // tensor_dim0[15:0]
        g1[2] = (int)((dim0 >> 16) | (1u << 16));        // tensor_dim0[31:16] | tensor_dim1=1
        g1[3] = (int)((unsigned)TILE_T << 16);           // tile_dim0 = TILE_T
        g1[4] = 1;                                       // tile_dim1 = 1
        g1[5] = TILE_T;                                  // tensor_dim0_stride
        g1[6] = 0;
        g1[7] = 0;

        v4i z4 = {0, 0, 0, 0};
#if __clang_major__ >= 23
        v8i z8 = {0, 0, 0, 0, 0, 0, 0, 0};
        __builtin_amdgcn_tensor_load_to_lds(g0, g1, z4, z4, z8, 0);
#else
        __builtin_amdgcn_tensor_load_to_lds(g0, g1, z4, z4, 0);
#endif
        __builtin_amdgcn_s_wait_tensorcnt(0);
    }
    __syncthreads();
#else
    // Fallback path (no TDM builtin): cooperative LDS fill.
    for (int i = threadIdx.x; i < TILE_T; i += blockDim.x) {
        int t = tstart + i;
        smod[i] = (t < T_LEN) ? mod[t] : 0.0f;
    }
    __syncthreads();
#endif

    // The TDM descriptor carries the LDS address opaquely, so the compiler
    // sees no store to smod and would fold the LDS load to undef.  Escape the
    // address into an asm memory clobber so the ds_load must be emitted.
    asm volatile("" :: "r"(&smod[0]) : "memory");

    const int t4 = tstart + (int)threadIdx.x * 4;
    if (t4 < T_LEN) {                       // T_LEN % 4 == 0 -> full b128 ok
        v4f m = *(const v4f*)(smod + threadIdx.x * 4);
        const int row0 = blockIdx.y * ROWS;
        for (int r = 0; r < ROWS; ++r) {
            size_t off = (size_t)(row0 + r) * T_LEN + (size_t)t4;
            v4f xv = __builtin_nontemporal_load((const v4f*)(x + off));
            v4f ov = xv * m;
            __builtin_nontemporal_store(ov, (v4f*)(out + off));
        }
    }
}

// ---------------------------------------------------------------------------
// Pass 1: per-chunk biquad with zero initial y-state (input-driven response).
// One thread per chunk; latency-bound dependent FMA chain of length L_CHUNK.
// ---------------------------------------------------------------------------
__global__ void iir_partials_kernel(const float* __restrict__ noise,
                                    float* __restrict__ partials,
                                    float b0, float b1, float b2,
                                    float a1, float a2)
{
    int k = blockIdx.x * blockDim.x + threadIdx.x;
    if (k >= K_CHUNKS) return;
    int s = k * L_CHUNK;
    // x1/x2 are *known* inputs at the chunk boundary (not part of the
    // propagated unknown state, which is only (y1, y2)).
    float x1 = (s >= 1) ? 0.5f * noise[s - 1] : 0.0f;
    float x2 = (s >= 2) ? 0.5f * noise[s - 2] : 0.0f;
    float y1 = 0.0f, y2 = 0.0f;
    for (int i = 0; i < L_CHUNK; ++i) {
        float xn = 0.5f * noise[s + i];
        float t  = b0 * xn + b1 * x1 + b2 * x2;   // independent of y chain
        float y  = t - a1 * y1 - a2 * y2;         // 2-FMA critical path
        x2 = x1; x1 = xn; y2 = y1; y1 = y;
    }
    partials[2 * k]     = y1;
    partials[2 * k + 1] = y2;
}

// ---------------------------------------------------------------------------
// Pass 2: serial chain of K 2x2 affine maps: s_k = M*s_{k-1} + v_k,
// M = A^L (host-computed in double). Stores the start state of each chunk.
// ---------------------------------------------------------------------------
__global__ void iir_fixup_kernel(const float* __restrict__ partials,
                                 float* __restrict__ starts,
                                 float m00, float m01, float m10, float m11)
{
    if (threadIdx.x != 0 || blockIdx.x != 0) return;
    float y1 = 0.0f, y2 = 0.0f;
    for (int k = 0; k < K_CHUNKS; ++k) {
        starts[2 * k]     = y1;
        starts[2 * k + 1] = y2;
        float v1 = partials[2 * k];
        float v2 = partials[2 * k + 1];
        float n1 = v1 + m00 * y1 + m01 * y2;
        float n2 = v2 + m10 * y1 + m11 * y2;
        y1 = n1; y2 = n2;
    }
}

// ---------------------------------------------------------------------------
// Pass 3: replay each chunk with its correct start state; write mod = y + 1.
// ---------------------------------------------------------------------------
__global__ void iir_final_kernel(const float* __restrict__ noise,
                                 const float* __restrict__ starts,
                                 float* __restrict__ mod,
                                 float b0, float b1, float b2,
                                 float a1, float a2)
{
    int k = blockIdx.x * blockDim.x + threadIdx.x;
    if (k >= K_CHUNKS) return;
    int s = k * L_CHUNK;
    float x1 = (s >= 1) ? 0.5f * noise[s - 1] : 0.0f;
    float x2 = (s >= 2) ? 0.5f * noise[s - 2] : 0.0f;
    float y1 = starts[2 * k];
    float y2 = starts[2 * k + 1];
    for (int i = 0; i < L_CHUNK; ++i) {
        float xn = 0.5f * noise[s + i];
        float t  = b0 * xn + b1 * x1 + b2 * x2;
        float y  = t - a1 * y1 - a2 * y2;
        mod[s + i] = y + 1.0f;                    // + OFFSET
        x2 = x1; x1 = xn; y2 = y1; y1 = y;
    }
}

// ---------------------------------------------------------------------------
// Host side
// ---------------------------------------------------------------------------
static inline void mat2_mul(double r[4], const double p[4], const double q[4]) {
    double t0 = p[0] * q[0] + p[1] * q[2];
    double t1 = p[0] * q[1] + p[1] * q[3];
    double t2 = p[2] * q[0] + p[3] * q[2];
    double t3 = p[2] * q[1] + p[3] * q[3];
    r[0] = t0; r[1] = t1; r[2] = t2; r[3] = t3;
}

extern "C" void kernel_launch(void* const* d_in, const int* in_sizes, int n_in,
                              void* d_out, int out_size, void* d_ws, size_t ws_size,
                              hipStream_t stream)
{
    (void)in_sizes; (void)n_in; (void)out_size; (void)ws_size;

    const float* x     = (const float*)d_in[0];   // (256, 160000) f32
    const float* noise = (const float*)d_in[1];   // (1, 160000)  f32
    float* out = (float*)d_out;

    float* mod      = (float*)d_ws;               // 160000 floats
    float* partials = mod + T_LEN;                // 2*K floats
    float* starts   = partials + 2 * K_CHUNKS;    // 2*K floats

    // torchaudio lowpass_biquad coefficients (double, rounded to f32 to match
    // the reference's np.float32 casts).
    const double PI   = 3.14159265358979323846;
    const double cutoff = 0.5 * 2.0;              // 0.5 * SPEED = 1.0 Hz
    const double w0   = 2.0 * PI * cutoff / 16000.0;
    const double alpha = sin(w0) / (2.0 * 0.707);
    const double cosw = cos(w0);
    const double a0   = 1.0 + alpha;
    const float b0f = (float)((1.0 - cosw) / 2.0 / a0);
    const float b1f = (float)((1.0 - cosw) / a0);
    const float b2f = b0f;
    const float a1f = (float)(-2.0 * cosw / a0);
    const float a2f = (float)((1.0 - alpha) / a0);

    // M = A^L_CHUNK in double, A = [[-a1, -a2], [1, 0]] (f32-rounded coeffs).
    double A[4] = { -(double)a1f, -(double)a2f, 1.0, 0.0 };
    double M[4] = { 1.0, 0.0, 0.0, 1.0 };
    double P[4] = { A[0], A[1], A[2], A[3] };
    int e = L_CHUNK;
    while (e > 0) {
        if (e & 1) mat2_mul(M, M, P);
        mat2_mul(P, P, P);
        e >>= 1;
    }

    const int cgrid = (K_CHUNKS + 31) / 32;
    iir_partials_kernel<<<cgrid, 32, 0, stream>>>(noise, partials,
                                                  b0f, b1f, b2f, a1f, a2f);
    iir_fixup_kernel<<<1, 32, 0, stream>>>(partials, starts,
                                           (float)M[0], (float)M[1],
                                           (float)M[2], (float)M[3]);
    iir_final_kernel<<<cgrid, 32, 0, stream>>>(noise, starts, mod,
                                               b0f, b1f, b2f, a1f, a2f);

    dim3 mgrid((T_LEN + TILE_T - 1) / TILE_T, N_CH / ROWS);
    modmul_kernel<<<mgrid, 256, 0, stream>>>(x, mod, out);
}